// Sequence2_34754875359850
// MI455X (gfx1250) — compile-verified
//
#include <hip/hip_runtime.h>

// ---------------------------------------------------------------------------
// Persistent 4-layer LSTM stack for MI455X (gfx1250), wave32, f32 WMMA.
// B=2048 batch rows, T=256 steps + 1 extra step. 64 WGs x 128 threads,
// 32 batch rows per WG. All weights + recurrent state resident in LDS.
// ---------------------------------------------------------------------------

typedef __attribute__((ext_vector_type(2))) float v2f;
typedef __attribute__((ext_vector_type(4))) float v4f;
typedef __attribute__((ext_vector_type(8))) float v8f;

// ---- problem constants ----
#define BATCH   2048
#define TSTEPS  256
#define HID     51
#define ROWS_WG 32           // batch rows per workgroup
#define NWG     (BATCH / ROWS_WG)

// ---- LDS layout (float offsets), all 16B aligned ----
// W stored transposed [n][k], gate-interleaved n = 4*j + g.
#define W1_STR   54          // K pad of 51 -> 52, stride 54 (bank-friendly)
#define W23_STR  106         // K pad of 102 -> 104, stride 106
#define IN1_STR  54
#define IN23_STR 106
#define C3_STR   54
#define GSTR     212         // gates buffer stride (208 cols + pad)

#define W1_OFF   0                         // 208*54  = 11232
#define W2_OFF   (W1_OFF + 208*54)         // 208*106 = 22048
#define W3_OFF   (W2_OFF + 208*106)
#define W4_OFF   (W3_OFF + 208*106)        // 204 -> pad 208
#define WHH4_OFF (W4_OFF + 208)            // 4
#define B4_OFF   (WHH4_OFF + 4)            // 4
#define B1_OFF   (B4_OFF + 4)              // 208
#define B2_OFF   (B1_OFF + 208)
#define B3_OFF   (B2_OFF + 208)
#define WIH1_OFF (B3_OFF + 208)            // 208
#define IN1_OFF  (WIH1_OFF + 208)          // 32*54  = 1728
#define IN2_OFF  (IN1_OFF + 32*IN1_STR)    // 32*106 = 3392
#define IN3_OFF  (IN2_OFF + 32*IN23_STR)
#define C3_OFF   (IN3_OFF + 32*IN23_STR)   // 32*54  = 1728
#define GAT_OFF  (C3_OFF + 32*C3_STR)      // 32*212 = 6784
#define XB_OFF   (GAT_OFF + 32*GSTR)       // 32
#define SMEM_FLOATS (XB_OFF + 32)
#define SMEM_BYTES  (SMEM_FLOATS * 4)      // 293,728 B < 320 KB

__device__ __forceinline__ float sigf(float x) {
    return 1.0f / (1.0f + __expf(-x));
}
__device__ __forceinline__ float tanhfast(float x) {
    float e = __expf(-2.0f * fabsf(x));     // e in (0,1], no overflow
    float r = (1.0f - e) / (1.0f + e);
    return copysignf(r, x);
}

// One wave computes its 16x16 M-tile against N-tiles [ntB, ntE) with K steps
// of 4 using V_WMMA_F32_16X16X4_F32.  A = in[row][k] (LDS, stride INS),
// B = W[n][k] transposed (LDS, stride WS).  Result -> gates LDS buffer.
// mt/ntB/ntE are wave-uniform (pre-scalarized by the caller via
// readfirstlane) so this lowers to a scalar loop with EXEC all-ones, as
// WMMA requires.
template <int INS, int K, int WS>
__device__ __forceinline__ void wmma_tiles(const float* __restrict__ inb,
                                           const float* __restrict__ wb,
                                           float* __restrict__ gat,
                                           int mt, int ntB, int ntE, int lane) {
    const int r  = lane & 15;
    const int hi = lane >> 4;
    // A fragment layout (ISA 7.12.2, 32-bit A 16x4): lanes 0-15 hold K={0,1},
    // lanes 16-31 hold K={2,3}, row = lane%16, VGPR0=lower K.
    const float* arow = inb + (mt * 16 + r) * INS + 2 * hi;
    for (int nt = ntB; nt < ntE; ++nt) {
        const float* brow = wb + (nt * 16 + r) * WS + 2 * hi;  // B[k][n]=W^T
        v8f acc = {0.f, 0.f, 0.f, 0.f, 0.f, 0.f, 0.f, 0.f};
        #pragma unroll
        for (int k = 0; k < K; k += 4) {
            v2f a = *(const v2f*)(arow + k);
            v2f b = *(const v2f*)(brow + k);
            acc = __builtin_amdgcn_wmma_f32_16x16x4_f32(
                false, a, false, b, (short)0, acc, false, false);
        }
        // C/D layout: VGPR v, lanes 0-15: M=v, N=lane; lanes 16-31: M=8+v.
        float* g0 = gat + (mt * 16 + 8 * hi) * GSTR + nt * 16 + r;
        #pragma unroll
        for (int v = 0; v < 8; ++v) g0[v * GSTR] = acc[v];
    }
}

// Activation: one (row, hidden-unit) per work item; gate order interleaved so
// (i,f,g,o) is one contiguous b128 in the gates buffer.
template <bool L1>
__device__ __forceinline__ void act_layer(const float* __restrict__ gat,
                                          const float* __restrict__ bS,
                                          const float* __restrict__ wih,
                                          const float* __restrict__ xbuf,
                                          float* __restrict__ cArr, int cStride,
                                          float* __restrict__ hArr, int hStride,
                                          int hOff, int tid) {
    for (int it = 0; it < 13; ++it) {            // 32*51 = 1632 items / 128
        int idx = tid + it * 128;
        if (idx < 32 * HID) {
            int row = idx / HID;
            int j   = idx - row * HID;
            v4f g4 = *(const v4f*)(gat + row * GSTR + 4 * j);
            v4f bb = *(const v4f*)(bS + 4 * j);
            float gi = g4.x + bb.x, gf = g4.y + bb.y;
            float gg = g4.z + bb.z, go = g4.w + bb.w;
            if (L1) {                            // layer1: x (scalar) * w_ih1
                v4f wv = *(const v4f*)(wih + 4 * j);
                float xv = xbuf[row];
                gi = fmaf(xv, wv.x, gi); gf = fmaf(xv, wv.y, gf);
                gg = fmaf(xv, wv.z, gg); go = fmaf(xv, wv.w, go);
            }
            float cOld = cArr[row * cStride + j];
            float cN = sigf(gf) * cOld + sigf(gi) * tanhfast(gg);
            cArr[row * cStride + j] = cN;
            hArr[row * hStride + hOff + j] = sigf(go) * tanhfast(cN);
        }
    }
}

__global__ void __launch_bounds__(128, 1)
lstm_stack_kernel(const float* __restrict__ x,
                  const float* __restrict__ wih1, const float* __restrict__ whh1,
                  const float* __restrict__ bih1, const float* __restrict__ bhh1,
                  const float* __restrict__ wih2, const float* __restrict__ whh2,
                  const float* __restrict__ bih2, const float* __restrict__ bhh2,
                  const float* __restrict__ wih3, const float* __restrict__ whh3,
                  const float* __restrict__ bih3, const float* __restrict__ bhh3,
                  const float* __restrict__ wih4, const float* __restrict__ whh4,
                  const float* __restrict__ bih4, const float* __restrict__ bhh4,
                  float* __restrict__ out) {
    extern __shared__ float sm[];
    float* W1   = sm + W1_OFF;
    float* W2   = sm + W2_OFF;
    float* W3   = sm + W3_OFF;
    float* w4s  = sm + W4_OFF;
    float* whh4s = sm + WHH4_OFF;
    float* b4s  = sm + B4_OFF;
    float* b1s  = sm + B1_OFF;
    float* b2s  = sm + B2_OFF;
    float* b3s  = sm + B3_OFF;
    float* wih1s = sm + WIH1_OFF;
    float* in1  = sm + IN1_OFF;   // h1           [32][54]
    float* in2  = sm + IN2_OFF;   // [c1 | h2]    [32][106]
    float* in3  = sm + IN3_OFF;   // [c2 | h3]    [32][106]
    float* c3b  = sm + C3_OFF;    // c3           [32][54]
    float* gat  = sm + GAT_OFF;   // gates        [32][212]
    float* xb   = sm + XB_OFF;    // x_t per row  [32]

    const int tid = threadIdx.x;
    const int b0  = blockIdx.x * ROWS_WG;

    // ------------------ init: build permuted/transposed weights ------------
    // W1[n][k] = w_hh1[g*51+j][k], n = 4*j+g  (zero-padded)
    for (int idx = tid; idx < 208 * W1_STR; idx += 128) {
        int n = idx / W1_STR, k = idx - n * W1_STR;
        int j = n >> 2, g = n & 3;
        float v = 0.f;
        if (j < HID && k < HID) v = whh1[(g * HID + j) * HID + k];
        W1[idx] = v;
    }
    // W2[n][k]: k<51 -> w_ih2, 52<=k<103 -> w_hh2 (concat input [c1|h2])
    for (int idx = tid; idx < 208 * W23_STR; idx += 128) {
        int n = idx / W23_STR, k = idx - n * W23_STR;
        int j = n >> 2, g = n & 3;
        float v = 0.f;
        if (j < HID) {
            int rsrc = (g * HID + j) * HID;
            if (k < HID)                 v = wih2[rsrc + k];
            else if (k >= 52 && k < 103) v = whh2[rsrc + (k - 52)];
        }
        W2[idx] = v;
    }
    for (int idx = tid; idx < 208 * W23_STR; idx += 128) {
        int n = idx / W23_STR, k = idx - n * W23_STR;
        int j = n >> 2, g = n & 3;
        float v = 0.f;
        if (j < HID) {
            int rsrc = (g * HID + j) * HID;
            if (k < HID)                 v = wih3[rsrc + k];
            else if (k >= 52 && k < 103) v = whh3[rsrc + (k - 52)];
        }
        W3[idx] = v;
    }
    // interleaved biases and layer-1 input weight column
    for (int n = tid; n < 208; n += 128) {
        int j = n >> 2, g = n & 3;
        bool ok = j < HID;
        int s = g * HID + j;
        b1s[n]   = ok ? (bih1[s] + bhh1[s]) : 0.f;
        b2s[n]   = ok ? (bih2[s] + bhh2[s]) : 0.f;
        b3s[n]   = ok ? (bih3[s] + bhh3[s]) : 0.f;
        wih1s[n] = ok ? wih1[s] : 0.f;      // w_ih1 is (204,1) flat
    }
    for (int idx = tid; idx < 208; idx += 128)
        w4s[idx] = (idx < 4 * HID) ? wih4[idx] : 0.f;   // (4,51) row-major
    if (tid < 4) { whh4s[tid] = whh4[tid]; b4s[tid] = bih4[tid] + bhh4[tid]; }
    // zero recurrent state (in1,in2,in3,c3b are contiguous)
    for (int idx = tid; idx < (GAT_OFF - IN1_OFF); idx += 128) in1[idx] = 0.f;
    __syncthreads();

    const int lane = tid & 31;
    // Wave-uniform tile assignment: force into SGPRs so the tile loops are
    // scalar (s_cbranch) and EXEC stays trivially all-ones through the WMMAs.
    const int wave = __builtin_amdgcn_readfirstlane(tid >> 5);
    const int mt  = wave >> 1;               // M-tile (16 rows)
    const int nh  = wave & 1;                // N half
    const int ntB = nh ? 7 : 0, ntE = nh ? 13 : 7;
    const int row4 = tid >> 2, gg4 = tid & 3;

    float h4 = 0.f, c4 = 0.f;

    for (int t = 0; t <= TSTEPS; ++t) {
        if (t < TSTEPS && tid < ROWS_WG)
            xb[tid] = x[(b0 + tid) * TSTEPS + t];   // else xb holds c4 (extra step)

        // ---- layer 1: gates = h1 @ W1 (x-term folded into activation) ----
        wmma_tiles<IN1_STR, 52, W1_STR>(in1, W1, gat, mt, ntB, ntE, lane);
        __syncthreads();
        act_layer<true>(gat, b1s, wih1s, xb, in2, IN23_STR, in1, IN1_STR, 0, tid);
        __syncthreads();

        // ---- layer 2: gates = [c1|h2] @ W2 ----
        wmma_tiles<IN23_STR, 104, W23_STR>(in2, W2, gat, mt, ntB, ntE, lane);
        __syncthreads();
        act_layer<false>(gat, b2s, nullptr, nullptr, in3, IN23_STR, in2, IN23_STR, 52, tid);
        __syncthreads();

        // ---- layer 3: gates = [c2|h3] @ W3 ----
        wmma_tiles<IN23_STR, 104, W23_STR>(in3, W3, gat, mt, ntB, ntE, lane);
        __syncthreads();
        act_layer<false>(gat, b3s, nullptr, nullptr, c3b, C3_STR, in3, IN23_STR, 52, tid);
        __syncthreads();

        // ---- layer 4 (hidden=1): VALU, one (row,gate) per thread ----
        {
            float s = b4s[gg4] + h4 * whh4s[gg4];
            const float* crow = c3b + row4 * C3_STR;
            const float* wrow = w4s + gg4 * HID;
            #pragma unroll 3
            for (int k = 0; k < HID; ++k) s = fmaf(crow[k], wrow[k], s);
            int base = lane & ~3;                      // quad gather (wave32)
            float gi = __shfl(s, base + 0, 32);
            float gf = __shfl(s, base + 1, 32);
            float gz = __shfl(s, base + 2, 32);
            float go = __shfl(s, base + 3, 32);
            c4 = sigf(gf) * c4 + sigf(gi) * tanhfast(gz);
            h4 = sigf(go) * tanhfast(c4);
            if (t == TSTEPS - 1 && gg4 == 0) xb[row4] = c4;  // extra-step input
        }
        // no barrier needed: next phase (WMMA L1) touches none of layer-4's
        // buffers, and xb is re-read only after the post-WMMA barrier.
    }

    if (gg4 == 0) out[b0 + row4] = c4;
}

extern "C" void kernel_launch(void* const* d_in, const int* in_sizes, int n_in,
                              void* d_out, int out_size, void* d_ws, size_t ws_size,
                              hipStream_t stream) {
    const float* x    = (const float*)d_in[0];
    const float* wih1 = (const float*)d_in[1];
    const float* whh1 = (const float*)d_in[2];
    const float* bih1 = (const float*)d_in[3];
    const float* bhh1 = (const float*)d_in[4];
    const float* wih2 = (const float*)d_in[5];
    const float* whh2 = (const float*)d_in[6];
    const float* bih2 = (const float*)d_in[7];
    const float* bhh2 = (const float*)d_in[8];
    const float* wih3 = (const float*)d_in[9];
    const float* whh3 = (const float*)d_in[10];
    const float* bih3 = (const float*)d_in[11];
    const float* bhh3 = (const float*)d_in[12];
    const float* wih4 = (const float*)d_in[13];
    const float* whh4 = (const float*)d_in[14];
    const float* bih4 = (const float*)d_in[15];
    const float* bhh4 = (const float*)d_in[16];
    float* out = (float*)d_out;

    (void)hipFuncSetAttribute((const void*)lstm_stack_kernel,
                              hipFuncAttributeMaxDynamicSharedMemorySize,
                              SMEM_BYTES);
    lstm_stack_kernel<<<dim3(NWG), dim3(128), SMEM_BYTES, stream>>>(
        x, wih1, whh1, bih1, bhh1, wih2, whh2, bih2, bhh2,
        wih3, whh3, bih3, bhh3, wih4, whh4, bih4, bhh4, out);
}